// Criterion_57037165691508
// MI455X (gfx1250) — compile-verified
//
#include <hip/hip_runtime.h>
#include <math.h>

#define BB 4
#define VV 8192
#define FF 16384
#define NN 8192
#define EPSV 1e-3f
#define WEIGHT 1000.0f

typedef __attribute__((ext_vector_type(2))) float v2f;
typedef __attribute__((ext_vector_type(8))) float v8f;

// ---------------------------------------------------------------------------
// Phase 1: per-face centers (packed with |c|^2) and unit normals.
// centers4[b*FF+f] = {cx, cy, cz, |c|^2}, normals4[b*FF+f] = {nx, ny, nz, 0}
// ---------------------------------------------------------------------------
__global__ void faces_prep_kernel(const float* __restrict__ pos,
                                  const int* __restrict__ faces,
                                  float* __restrict__ centers4,
                                  float* __restrict__ normals4) {
  int gid = blockIdx.x * blockDim.x + threadIdx.x;
  if (gid >= BB * FF) return;
  int b = gid / FF;
  const int* fc = faces + (size_t)gid * 3;
  const float* pb = pos + (size_t)b * VV * 3;
  int i0 = fc[0], i1 = fc[1], i2 = fc[2];
  float ax = pb[i0 * 3 + 0], ay = pb[i0 * 3 + 1], az = pb[i0 * 3 + 2];
  float bx = pb[i1 * 3 + 0], by = pb[i1 * 3 + 1], bz = pb[i1 * 3 + 2];
  float cx = pb[i2 * 3 + 0], cy = pb[i2 * 3 + 1], cz = pb[i2 * 3 + 2];

  const float third = 1.0f / 3.0f;
  float mx = (ax + bx + cx) * third;
  float my = (ay + by + cy) * third;
  float mz = (az + bz + cz) * third;
  float c2 = mx * mx + my * my + mz * mz;

  float e1x = bx - ax, e1y = by - ay, e1z = bz - az;
  float e2x = cx - ax, e2y = cy - ay, e2z = cz - az;
  float nx = e1y * e2z - e1z * e2y;
  float ny = e1z * e2x - e1x * e2z;
  float nz = e1x * e2y - e1y * e2x;
  float inv = 1.0f / fmaxf(sqrtf(nx * nx + ny * ny + nz * nz), 1e-12f);

  float4 cv = make_float4(mx, my, mz, c2);
  float4 nv = make_float4(nx * inv, ny * inv, nz * inv, 0.0f);
  *(float4*)(centers4 + (size_t)gid * 4) = cv;
  *(float4*)(normals4 + (size_t)gid * 4) = nv;
}

// ---------------------------------------------------------------------------
// Phase 2: WMMA brute-force 1-NN + loss accumulation.
// score[f][n] = |c_f|^2 - 2 p_n . c_f  computed as A(16x4) x B(4x16) WMMA:
//   A row (face f)  = ( cx, cy, cz, |c|^2 )
//   B col (point n) = (-2px, -2py, -2pz, 1 )
// D layout: VGPR r, lanes 0-15 -> (face tile+r,  point lane)
//           VGPR r, lanes16-31 -> (face tile+8+r, point lane-16)
// so each lane tracks a per-point running (min, argmin) entirely in-register.
// ---------------------------------------------------------------------------
__global__ void nn_loss_kernel(const float* __restrict__ pred,
                               const float* __restrict__ centers4,
                               const float* __restrict__ normals4,
                               float* __restrict__ partials) {
  const int lane = threadIdx.x & 31;
  const int wave = threadIdx.x >> 5;
  const int blocksPerBatch = NN / 128;  // 64
  const int b = blockIdx.x / blocksPerBatch;
  const int pointBase = (blockIdx.x % blocksPerBatch) * 128 + wave * 16;
  const int laneN = lane & 15;
  const bool hi = lane >= 16;

  // Every lane loads the full point for column laneN (hi lanes need pz).
  const float* pp = pred + ((size_t)b * NN + pointBase + laneN) * 3;
  float px = pp[0], py = pp[1], pz = pp[2];

  // B-matrix fragment (4x16 f32, 2 VGPRs):
  //   VGPR0: lanes0-15 K=0 (-2px), lanes16-31 K=2 (-2pz)
  //   VGPR1: lanes0-15 K=1 (-2py), lanes16-31 K=3 (1.0)
  v2f bmat;
  bmat.x = hi ? (-2.0f * pz) : (-2.0f * px);
  bmat.y = hi ? 1.0f : (-2.0f * py);

  const float* cb = centers4 + (size_t)b * FF * 4;
  float bestV = 3.4e38f;
  int bestI = 0;

#pragma unroll 4
  for (int t = 0; t < FF; t += 16) {
    // A-matrix fragment (16x4 f32, 2 VGPRs):
    //   lanes0-15  : (K0,K1) = (cx,cy)   of face t+laneN
    //   lanes16-31 : (K2,K3) = (cz,|c|^2) of face t+laneN
    const v2f a = *(const v2f*)(cb + (size_t)(t + laneN) * 4 + (hi ? 2 : 0));
    v8f acc = {};
    acc = __builtin_amdgcn_wmma_f32_16x16x4_f32(
        /*neg_a=*/false, a, /*neg_b=*/false, bmat,
        /*c_mod=*/(short)0, acc, /*reuse_a=*/false, /*reuse_b=*/false);

    const int base = t + (hi ? 8 : 0);
#pragma unroll
    for (int r = 0; r < 8; ++r) {
      float v = acc[r];
      bool take = v < bestV;  // strict '<' keeps first (lowest-index) min
      bestI = take ? (base + r) : bestI;
      bestV = take ? v : bestV;
    }
  }

  // Lanes L and L^16 hold the same point's two face-subsets; merge them.
  float oV = __shfl_xor(bestV, 16, 32);
  int oI = __shfl_xor(bestI, 16, 32);
  if (oV < bestV || (oV == bestV && oI < bestI)) { bestV = oV; bestI = oI; }

  float i3 = 0.0f, cnt = 0.0f;
  if (!hi) {
    float4 c = *(const float4*)(cb + (size_t)bestI * 4);
    float4 nrm = *(const float4*)(normals4 + ((size_t)b * FF + bestI) * 4);
    float dist = (px - c.x) * nrm.x + (py - c.y) * nrm.y + (pz - c.z) * nrm.z;
    float interp = fmaxf(EPSV - dist, 0.0f);
    i3 = interp * interp * interp;
    cnt = (interp > 0.0f) ? 1.0f : 0.0f;
  }

  __shared__ float sI[256];
  __shared__ float sC[256];
  sI[threadIdx.x] = i3;
  sC[threadIdx.x] = cnt;
  __syncthreads();
#pragma unroll
  for (int s = 128; s > 0; s >>= 1) {
    if ((int)threadIdx.x < s) {
      sI[threadIdx.x] += sI[threadIdx.x + s];
      sC[threadIdx.x] += sC[threadIdx.x + s];
    }
    __syncthreads();
  }
  if (threadIdx.x == 0) {
    partials[blockIdx.x * 2 + 0] = sI[0];
    partials[blockIdx.x * 2 + 1] = sC[0];
  }
}

// ---------------------------------------------------------------------------
// Phase 3: fixed-order reduction of per-block partials -> (loss, perc)
// ---------------------------------------------------------------------------
__global__ void finalize_kernel(const float* __restrict__ partials, int nBlocks,
                                float* __restrict__ out) {
  __shared__ float sI[256];
  __shared__ float sC[256];
  int t = threadIdx.x;
  float i3 = 0.0f, c = 0.0f;
  if (t < nBlocks) {
    i3 = partials[t * 2 + 0];
    c = partials[t * 2 + 1];
  }
  sI[t] = i3;
  sC[t] = c;
  __syncthreads();
#pragma unroll
  for (int s = 128; s > 0; s >>= 1) {
    if (t < s) {
      sI[t] += sI[t + s];
      sC[t] += sC[t + s];
    }
    __syncthreads();
  }
  if (t == 0) {
    out[0] = sI[0] / (float)BB * WEIGHT;         // loss
    out[1] = sC[0] / (float)((size_t)BB * NN);   // perc
  }
}

extern "C" void kernel_launch(void* const* d_in, const int* in_sizes, int n_in,
                              void* d_out, int out_size, void* d_ws, size_t ws_size,
                              hipStream_t stream) {
  const float* pred = (const float*)d_in[0];   // [B,N,3] f32
  const float* pos = (const float*)d_in[1];    // [B,V,3] f32
  const int* faces = (const int*)d_in[2];      // [B,F,3] i32

  float* centers4 = (float*)d_ws;                          // B*F*4 floats (1 MB)
  float* normals4 = centers4 + (size_t)BB * FF * 4;        // B*F*4 floats (1 MB)
  float* partials = normals4 + (size_t)BB * FF * 4;        // 2 * nBlocks2 floats

  faces_prep_kernel<<<(BB * FF + 255) / 256, 256, 0, stream>>>(pos, faces,
                                                               centers4, normals4);
  const int nBlocks2 = BB * (NN / 128);  // 256
  nn_loss_kernel<<<nBlocks2, 256, 0, stream>>>(pred, centers4, normals4, partials);
  finalize_kernel<<<1, 256, 0, stream>>>(partials, nBlocks2, (float*)d_out);
}